// TopologyAnalyzer_3994319585572
// MI455X (gfx1250) — compile-verified
//
#include <hip/hip_runtime.h>
#include <math.h>

typedef __attribute__((ext_vector_type(2))) float v2f;
typedef __attribute__((ext_vector_type(8))) float v8f;

#define WAVES_PER_BLOCK 4
#define SAMPLES_PER_BLOCK 64

// Fused Phi(g) = 0.5*(1+erf(g/sqrt2)) and pdf(g) = exp(-g^2/2)/sqrt(2pi).
// Key identity: the erf approximation's exp(-(g/sqrt2)^2) IS exp(-g^2/2),
// so one v_exp_f32 serves both. Branchless (A&S 7.1.26, |err| < 1.5e-7).
__device__ __forceinline__ void phi_pdf(float g, float& Phi, float& pdf) {
    float x  = g * 0.70710678118654752440f;
    float ax = fabsf(x);
    float t  = __builtin_amdgcn_rcpf(fmaf(0.3275911f, ax, 1.0f));
    float p  = fmaf(t, 1.061405429f, -1.453152027f);
    p = fmaf(t, p, 1.421413741f);
    p = fmaf(t, p, -0.284496736f);
    p = fmaf(t, p, 0.254829592f);
    p = p * t;
    float e  = __expf(-ax * ax);                    // = exp(-g^2/2)
    float er = copysignf(fmaf(-p, e, 1.0f), g);     // erf(g/sqrt2)
    Phi = 0.5f * (1.0f + er);
    pdf = 0.39894228040143267794f * e;
}

__device__ __forceinline__ v8f wmma4(v2f a, v2f b, v8f c) {
    return __builtin_amdgcn_wmma_f32_16x16x4_f32(false, a, false, b, (short)0, c,
                                                 false, false);
}

__global__ __launch_bounds__(128) void topo_catastrophe_kernel(
    const float* __restrict__ pillars, const float* __restrict__ W1,
    const float* __restrict__ b1, const float* __restrict__ W2,
    const float* __restrict__ b2, const float* __restrict__ lng,
    const float* __restrict__ lnb, const float* __restrict__ Wc,
    const float* __restrict__ bc, float* __restrict__ out, int nSamples) {
    // ---- LDS ----
    __shared__ float s_w1p[4 * 64 * 2];               // W1 K-pairs [kp][j][2]; K5=b1
    __shared__ float s_bvg[64 * 16 * 2];              // interleaved B rows (V,grad)
    __shared__ float s_btr[32 * 16 * 2];              // trace B, K-pairs
    __shared__ float s_praw[WAVES_PER_BLOCK][80];     // raw pillars (16 x 5)
    __shared__ float s_U[WAVES_PER_BLOCK][16 * 128];  // activation staging
    __shared__ float s_red[WAVES_PER_BLOCK][16 * 8];  // [V', grad0..4, tr']

    const int tid  = threadIdx.x;
    const int lane = tid & 31;
    const int wv   = tid >> 5;
    const int ln15 = lane & 15;  // A row / B,C column
    const int lh   = lane >> 4;  // lane half

    // ---- stage weights (each slot written exactly once) ----
    for (int idx = tid; idx < 512; idx += 128) {
        int k = idx >> 6, j = idx & 63;
        float v = (k < 5) ? W1[k * 64 + j] : ((k == 5) ? b1[j] : 0.0f);
        s_w1p[((k >> 1) * 64 + j) * 2 + (k & 1)] = v;  // pair-interleaved
    }
    if (tid < 64) {
        int j = tid;
        float w2 = W2[j];
        float ssum = 0.0f;
#pragma unroll
        for (int i = 0; i < 5; i++) { float w = W1[i * 64 + j]; ssum += w * w; }
        float w2s = w2 * ssum;
#pragma unroll
        for (int n = 0; n < 16; n++) {
            int base = (j * 16 + n) * 2;
            s_bvg[base + 0] = (n == 0) ? w2 : 0.0f;                          // gelu row
            s_bvg[base + 1] = (n >= 1 && n <= 5) ? w2 * W1[(n - 1) * 64 + j]
                                                 : 0.0f;                     // gelu' row
            s_btr[((j >> 1) * 16 + n) * 2 + (j & 1)] = (n == 6) ? w2s : 0.0f;
        }
    }
    __syncthreads();

    // ---- per-wave: cooperative load of 16 samples x 5 pillars ----
    const int sbase = (blockIdx.x * WAVES_PER_BLOCK + wv) * 16;
    const int gbase = sbase * 5;
    const int total = nSamples * 5;
    for (int i = lane; i < 80; i += 32) {
        int gi = gbase + i;
        s_praw[wv][i] = (gi < total) ? pillars[gi] : 0.0f;
    }
    // same-wave DS ops are in-order: cross-lane LDS RAW below is safe.

    // ---- per-lane: clipped p and layernorm x for row ln15 ----
    float pc[5], xx[5];
    float mu = 0.0f, sq = 0.0f;
#pragma unroll
    for (int i = 0; i < 5; i++) {
        float v = s_praw[wv][ln15 * 5 + i];
        v = __builtin_isnan(v) ? 0.5f : v;
        v = fminf(fmaxf(v, 0.0f), 1.0f);
        pc[i] = v; mu += v; sq += v * v;
    }
    mu *= 0.2f;
    float var  = sq * 0.2f - mu * mu;   // biased variance (torch LayerNorm)
    float rstd = rsqrtf(var + 1e-5f);
#pragma unroll
    for (int i = 0; i < 5; i++) xx[i] = (pc[i] - mu) * rstd * lng[i] + lnb[i];

    // ---- A fragments; K=5 carries 1.0 so W1-row5(=b1) adds the bias ----
    v2f ap0, ap1, ax0, ax1;
    if (lh == 0) {
        ap0.x = pc[0]; ap0.y = pc[1]; ap1.x = pc[4]; ap1.y = 1.0f;
        ax0.x = xx[0]; ax0.y = xx[1]; ax1.x = xx[4]; ax1.y = 1.0f;
    } else {
        ap0.x = pc[2]; ap0.y = pc[3]; ap1.x = 0.0f; ap1.y = 0.0f;
        ax0.x = xx[2]; ax0.y = xx[3]; ax1.x = 0.0f; ax1.y = 0.0f;
    }

    const v8f zero8 = {0.f, 0.f, 0.f, 0.f, 0.f, 0.f, 0.f, 0.f};

    // ---- GEMM1a: G[16x64] = [Pc|1] @ [W1;b1] ----
    v8f accG[4];
#pragma unroll
    for (int nt = 0; nt < 4; nt++) {
        int col = nt * 16 + ln15;
        v2f bA = *(const v2f*)&s_w1p[((0 * 2 + lh) * 64 + col) * 2];  // K pair 0/1
        v2f bB = *(const v2f*)&s_w1p[((1 * 2 + lh) * 64 + col) * 2];  // K pair 2/3
        v8f c = wmma4(ap0, bA, zero8);
        accG[nt] = wmma4(ap1, bB, c);
    }

    // ---- fused U1/U2: K-interleaved (2j = gelu, 2j+1 = gelu') ----
    // One rcp + one exp per element yields both gelu(g) and gelu'(g).
#pragma unroll
    for (int nt = 0; nt < 4; nt++)
#pragma unroll
        for (int v = 0; v < 8; v++) {
            int m = v + 8 * lh, j = nt * 16 + ln15;
            float g = accG[nt][v];
            float Phi, pdf;
            phi_pdf(g, Phi, pdf);
            v2f u; u.x = g * Phi; u.y = fmaf(g, pdf, Phi);
            *(v2f*)&s_U[wv][m * 128 + 2 * j] = u;
        }

    // ---- GEMM2-VG: C2 = Ucat[16x128] @ Bcat -> cols {0:V', 1..5:grad} ----
    v8f c2 = zero8;
#pragma unroll 4
    for (int t = 0; t < 32; t++) {
        int k0 = t * 4 + 2 * lh;
        v2f a = *(const v2f*)&s_U[wv][ln15 * 128 + k0];
        v2f b = *(const v2f*)&s_bvg[(k0 >> 1) * 32 + ln15 * 2];
        c2 = wmma4(a, b, c2);
    }

    // ---- GEMM1b: H[16x64] = [X|1] @ [W1;b1] ----
    v8f accH[4];
#pragma unroll
    for (int nt = 0; nt < 4; nt++) {
        int col = nt * 16 + ln15;
        v2f bA = *(const v2f*)&s_w1p[((0 * 2 + lh) * 64 + col) * 2];
        v2f bB = *(const v2f*)&s_w1p[((1 * 2 + lh) * 64 + col) * 2];
        v8f c = wmma4(ax0, bA, zero8);
        accH[nt] = wmma4(ax1, bB, c);
    }

    // ---- U3 = gelu''(H) = pdf(h)*(2-h^2): one exp per element ----
#pragma unroll
    for (int nt = 0; nt < 4; nt++)
#pragma unroll
        for (int v = 0; v < 8; v++) {
            int m = v + 8 * lh, j = nt * 16 + ln15;
            float h   = accH[nt][v];
            float pdf = 0.39894228040143267794f * __expf(-0.5f * h * h);
            s_U[wv][m * 128 + j] = pdf * (2.0f - h * h);
        }

    // ---- GEMM2-TR: C2 += U3[16x64] @ Btr -> col 6 = tr(H) ----
#pragma unroll 4
    for (int t = 0; t < 16; t++) {
        int k0 = t * 4 + 2 * lh;
        v2f a = *(const v2f*)&s_U[wv][ln15 * 128 + k0];
        v2f b = *(const v2f*)&s_btr[(k0 >> 1) * 32 + ln15 * 2];
        c2 = wmma4(a, b, c2);
    }

    // ---- spill C2 cols 0..7 per sample ----
#pragma unroll
    for (int v = 0; v < 8; v++) {
        int m = v + 8 * lh;
        if (ln15 < 8) s_red[wv][m * 8 + ln15] = c2[v];
    }

    // ---- final: feats -> L2 normalize -> classifier -> sigmoid -> clamp ----
    if (lane < 16) {
        int sidx = sbase + lane;
        if (sidx < nSamples) {
            float f[12];
#pragma unroll
            for (int i = 0; i < 5; i++) f[i] = s_praw[wv][lane * 5 + i];  // raw
#pragma unroll
            for (int i = 0; i < 5; i++) f[5 + i] = s_red[wv][lane * 8 + 1 + i];
            f[10] = s_red[wv][lane * 8 + 0] + b2[0];   // V
            f[11] = s_red[wv][lane * 8 + 6] * 0.2f;    // e_mean = tr(H)/5
            float ss = 0.0f;
#pragma unroll
            for (int i = 0; i < 12; i++) ss += f[i] * f[i];
            float inv = 1.0f / fmaxf(sqrtf(ss), 1e-12f);
            float z = bc[0];
#pragma unroll
            for (int i = 0; i < 12; i++) z += (f[i] * inv) * Wc[i];
            float pr = 1.0f / (1.0f + __expf(-z));
            out[sidx] = fminf(fmaxf(pr, 0.0f), 1.0f);
        }
    }
}

extern "C" void kernel_launch(void* const* d_in, const int* in_sizes, int n_in,
                              void* d_out, int out_size, void* d_ws, size_t ws_size,
                              hipStream_t stream) {
    (void)n_in; (void)out_size; (void)d_ws; (void)ws_size;
    const float* pillars = (const float*)d_in[0];
    const float* W1      = (const float*)d_in[1];
    const float* b1      = (const float*)d_in[2];
    const float* W2      = (const float*)d_in[3];
    const float* b2      = (const float*)d_in[4];
    const float* lng     = (const float*)d_in[5];
    const float* lnb     = (const float*)d_in[6];
    const float* Wc      = (const float*)d_in[7];
    const float* bc      = (const float*)d_in[8];
    float* out = (float*)d_out;

    int nSamples = in_sizes[0] / 5;
    int grid = (nSamples + SAMPLES_PER_BLOCK - 1) / SAMPLES_PER_BLOCK;
    topo_catastrophe_kernel<<<grid, 128, 0, stream>>>(
        pillars, W1, b1, W2, b2, lng, lnb, Wc, bc, out, nSamples);
}